// HierKVBank_42545946034830
// MI455X (gfx1250) — compile-verified
//
#include <hip/hip_runtime.h>
#include <math.h>

typedef __attribute__((ext_vector_type(2))) float v2f;
typedef __attribute__((ext_vector_type(8))) float v8f;

#define B_    16
#define T_    8192
#define C_    256
#define W0_   128
#define W1_   64
#define W2_   16
#define STR_  64
#define NCH_  126            // (T_-W0_)/STR_

#define OUT_K0    0
#define OUT_V0    524288
#define OUT_K1    1048576
#define OUT_V1    1310720
#define OUT_K2    1572864
#define OUT_V2    1638400
#define OUT_RECON 1703936

// ws layout (floats): [0..2015] stage1 per-(b,chunk) residual partials
//                     [2048..2559] stage2 per-(b,pair) residual partials
#define WS2_OFF 2048

__device__ __forceinline__ float wave_sum32(float v) {
    #pragma unroll
    for (int m = 16; m >= 1; m >>= 1) v += __shfl_xor(v, m, 32);
    return v;
}

// ---------------- kernel 1: copy K0/V0 (last 128 rows) ----------------
__global__ __launch_bounds__(256) void copy_tail(const float* __restrict__ K,
                                                 const float* __restrict__ V,
                                                 float* __restrict__ out) {
    int idx = blockIdx.x * 256 + threadIdx.x;           // 0 .. B*W0*C-1
    if (idx >= B_ * W0_ * C_) return;
    int c = idx & (C_ - 1);
    int r = (idx >> 8) & (W0_ - 1);
    int b = idx >> 15;                                  // / (W0_*C_)
    size_t src = ((size_t)b * T_ + (T_ - W0_) + r) * C_ + c;
    out[OUT_K0 + idx] = K[src];
    out[OUT_V0 + idx] = V[src];
}

// ---------------- kernel 2: stage-1 pooling over 64-row chunks ----------------
__global__ __launch_bounds__(256) void stage1(const float* __restrict__ K,
                                              const float* __restrict__ V,
                                              const float* __restrict__ q0,
                                              float* __restrict__ out,
                                              float* __restrict__ ws) {
    const int blk = blockIdx.x;                 // b*NCH_ + nn
    const int b   = blk / NCH_;
    const int nn  = blk % NCH_;
    const int tid  = threadIdx.x;               // 0..255, owns column c = tid
    const int lane = tid & 31;
    const int wv   = tid >> 5;

    const float* Kbase = K + ((size_t)b * T_ + (size_t)nn * STR_) * C_;
    const float* Vbase = V + ((size_t)b * T_ + (size_t)nn * STR_) * C_;

    __shared__ float s_part[64 * 8];
    __shared__ float s_logit[64];
    __shared__ float s_w[64];
    __shared__ float s_sk[256];
    __shared__ float s_red[8];

    // ---- pass 1: logits + first/second moments of K (single K read) ----
    const float qc = q0[tid];
    float sumK = 0.f, sumK2 = 0.f;
    for (int s = 0; s < STR_; ++s) {
        float kv = Kbase[s * C_ + tid];         // coalesced row load
        sumK  += kv;
        sumK2 += kv * kv;
        float p = wave_sum32(kv * qc);
        if (lane == 0) s_part[s * 8 + wv] = p;
    }
    __syncthreads();
    if (tid < 64) {
        float l = 0.f;
        #pragma unroll
        for (int w = 0; w < 8; ++w) l += s_part[tid * 8 + w];
        s_logit[tid] = l;
    }
    __syncthreads();
    if (tid < 64) {                              // softmax over 64 (tiny)
        float mx = -3.0e38f;
        for (int s = 0; s < 64; ++s) mx = fmaxf(mx, s_logit[s]);
        float den = 0.f;
        for (int s = 0; s < 64; ++s) den += __expf(s_logit[s] - mx);
        s_w[tid] = __expf(s_logit[tid] - mx) / den;
    }
    __syncthreads();

    // ---- pass 2: sk = w^T * Kc via V_WMMA_F32_16X16X4_F32 ----
    // A(16x4): A[m][k] = Kc[s0+k][c0+m]; B(4x16): w[s0+k] broadcast over N.
    // D[m][n] = sk[c0+m] (all n equal). 8 waves x 2 c-tiles = 256 columns.
    const int m_  = lane & 15;
    const int kk  = (lane >> 4) << 1;            // 0 or 2
    #pragma unroll
    for (int tci = 0; tci < 2; ++tci) {
        const int c0 = (wv + tci * 8) * 16;
        v8f acc = {};
        for (int s0 = 0; s0 < STR_; s0 += 4) {
            v2f a, bb;
            a.x  = Kbase[(s0 + kk)     * C_ + c0 + m_];   // L2-resident reread
            a.y  = Kbase[(s0 + kk + 1) * C_ + c0 + m_];
            bb.x = s_w[s0 + kk];
            bb.y = s_w[s0 + kk + 1];
            acc = __builtin_amdgcn_wmma_f32_16x16x4_f32(
                      false, a, false, bb, (short)0, acc, false, false);
        }
        // N=0 column lives in lanes 0 (M=0..7) and 16 (M=8..15)
        if (lane == 0) {
            #pragma unroll
            for (int j = 0; j < 8; ++j) s_sk[c0 + j] = acc[j];
        }
        if (lane == 16) {
            #pragma unroll
            for (int j = 0; j < 8; ++j) s_sk[c0 + 8 + j] = acc[j];
        }
    }
    __syncthreads();

    // ---- residual via moments: sum_s (K - sk)^2 = sumK2 - 2 sk sumK + 64 sk^2
    const float skc = s_sk[tid];
    float res = sumK2 - 2.f * skc * sumK + 64.f * skc * skc;
    res = wave_sum32(res);
    if (lane == 0) s_red[wv] = res;
    __syncthreads();
    if (tid == 0) {
        float t = 0.f;
        #pragma unroll
        for (int i = 0; i < 8; ++i) t += s_red[i];
        ws[blk] = t;                             // deterministic per-chunk partial
    }

    // ---- outputs for last W1 chunks only (V touched only here) ----
    if (nn >= NCH_ - W1_) {
        const int j = nn - (NCH_ - W1_);
        const size_t o = ((size_t)b * W1_ + j) * C_ + tid;
        out[OUT_K1 + o] = skc;
        float sv = 0.f;
        for (int s = 0; s < STR_; ++s) {
            if (s + 8 < STR_) __builtin_prefetch(&Vbase[(s + 8) * C_ + tid], 0, 1);
            sv += s_w[s] * Vbase[s * C_ + tid];
        }
        out[OUT_V1 + o] = sv;
    }
}

// ---------------- kernel 3: stage-2 pooling over pairs of K1 rows ----------------
__global__ __launch_bounds__(256) void stage2(const float* __restrict__ q1,
                                              float* __restrict__ out,
                                              float* __restrict__ ws) {
    const int blk = blockIdx.x;                  // b*32 + n2
    const int b   = blk >> 5;
    const int n2  = blk & 31;
    const int tid  = threadIdx.x;
    const int lane = tid & 31;
    const int wv   = tid >> 5;

    __shared__ float s_p0[8], s_p1[8], s_red[8];

    const float* K1 = out + OUT_K1;
    const float* V1 = out + OUT_V1;
    const size_t r0 = ((size_t)b * W1_ + 2 * n2) * C_;

    const float k0 = K1[r0 + tid];
    const float k1 = K1[r0 + C_ + tid];
    const float qc = q1[tid];

    float p0 = wave_sum32(k0 * qc);
    float p1 = wave_sum32(k1 * qc);
    if (lane == 0) { s_p0[wv] = p0; s_p1[wv] = p1; }
    __syncthreads();
    float l0 = 0.f, l1 = 0.f;
    #pragma unroll
    for (int i = 0; i < 8; ++i) { l0 += s_p0[i]; l1 += s_p1[i]; }
    const float mx = fmaxf(l0, l1);
    const float e0 = __expf(l0 - mx), e1 = __expf(l1 - mx);
    const float inv = 1.f / (e0 + e1);
    const float w0 = e0 * inv, w1 = e1 * inv;

    const float sk = w0 * k0 + w1 * k1;
    float res = (k0 - sk) * (k0 - sk) + (k1 - sk) * (k1 - sk);
    res = wave_sum32(res);
    if (lane == 0) s_red[wv] = res;
    __syncthreads();
    if (tid == 0) {
        float t = 0.f;
        #pragma unroll
        for (int i = 0; i < 8; ++i) t += s_red[i];
        ws[WS2_OFF + blk] = t;
    }

    if (n2 >= 32 - W2_) {                        // last 16 pairs -> K2/V2
        const int j = n2 - (32 - W2_);
        const size_t o = ((size_t)b * W2_ + j) * C_ + tid;
        out[OUT_K2 + o] = sk;
        const float v0 = V1[r0 + tid];
        const float v1v = V1[r0 + C_ + tid];
        out[OUT_V2 + o] = w0 * v0 + w1 * v1v;
    }
}

// ---------------- kernel 4: recon scalar ----------------
__global__ __launch_bounds__(256) void reduce_recon(float* __restrict__ out,
                                                    const float* __restrict__ ws) {
    const int tid = threadIdx.x;
    const int lane = tid & 31;
    const int wv = tid >> 5;
    // recon = sum(rl)/126 + 0.5*sum(rl2)
    // rl[nn]  = sum_{b,s,c} P / (B*64*C)   -> scale1 = 1/(126*B*64*C)
    // rl2[n2] = sum_{b,s,c} P2 / (B*2*C)   -> scale2 = 0.5/(B*2*C)
    const float scale1 = 1.0f / (126.0f * 16.0f * 64.0f * 256.0f);
    const float scale2 = 0.5f / (16.0f * 2.0f * 256.0f);
    float a = 0.f, c = 0.f;
    for (int i = tid; i < B_ * NCH_; i += 256) a += ws[i];
    for (int i = tid; i < B_ * 32; i += 256)   c += ws[WS2_OFF + i];
    float v = a * scale1 + c * scale2;
    v = wave_sum32(v);
    __shared__ float s[8];
    if (lane == 0) s[wv] = v;
    __syncthreads();
    if (tid == 0) {
        float t = 0.f;
        #pragma unroll
        for (int i = 0; i < 8; ++i) t += s[i];
        out[OUT_RECON] = t;
    }
}

extern "C" void kernel_launch(void* const* d_in, const int* in_sizes, int n_in,
                              void* d_out, int out_size, void* d_ws, size_t ws_size,
                              hipStream_t stream) {
    const float* K  = (const float*)d_in[0];
    const float* V  = (const float*)d_in[1];
    const float* q0 = (const float*)d_in[2];
    const float* q1 = (const float*)d_in[3];
    // d_in[4] = stride (==64), baked in at compile time
    float* out = (float*)d_out;
    float* ws  = (float*)d_ws;

    copy_tail  <<<(B_ * W0_ * C_ + 255) / 256, 256, 0, stream>>>(K, V, out);
    stage1     <<<B_ * NCH_,                 256, 0, stream>>>(K, V, q0, out, ws);
    stage2     <<<B_ * 32,                   256, 0, stream>>>(q1, out, ws);
    reduce_recon<<<1,                        256, 0, stream>>>(out, ws);
}